// Diag_72559177499246
// MI455X (gfx1250) — compile-verified
//
#include <hip/hip_runtime.h>

// Diag-scale: out_real = x_real * exp(betas), out_imag = x_imag * exp(betas),
// scale broadcast over the 256 (b,c) slices. Pure streaming / HBM-bound:
// ~268 MB of traffic -> ~11.5 us floor at 23.3 TB/s on MI455X.

typedef __attribute__((ext_vector_type(4))) float v4f;   // native vector: OK for NT builtins

#define TPB      256                 // 8 wave32s per block
#define VEC      4                   // floats per thread per slice
#define DIM      65536               // H*W
#define DIMV     (DIM / VEC)         // 16384 v4f per slice
#define BC       256                 // B*C slices
#define SLICES   8                   // slices per block (y-dim)
#define NELEMV   ((size_t)BC * DIMV) // v4f count per tensor

__global__ __launch_bounds__(TPB) void diag_scale_kernel(
    const float* __restrict__ x_real,
    const float* __restrict__ x_imag,
    const float* __restrict__ betas,
    float* __restrict__ out)
{
    __shared__ float sbeta[TPB * VEC];   // 4 KB betas tile for this x-block

    const int tid = threadIdx.x;

    // ---- Stage this block's betas tile into LDS with the CDNA5 async copy.
    // GVS mode: mem_addr = SADDR(betas) + VADDR(32-bit byte offset).
    // VDST VGPR holds the LDS byte address (low 32 bits of a generic LDS
    // pointer are the LDS offset per the gfx1250 aperture mapping).
    {
        unsigned lds_addr = (unsigned)(uintptr_t)(&sbeta[0]) + (unsigned)tid * 16u;
        unsigned goff     = (blockIdx.x * (TPB * VEC) + tid * VEC) * (unsigned)sizeof(float);
        asm volatile("global_load_async_to_lds_b128 %0, %1, %2 offset:0"
                     :
                     : "v"(lds_addr), "v"(goff), "s"(betas)
                     : "memory");
        asm volatile("s_wait_asynccnt 0x0" ::: "memory");
    }
    // Each thread reads back exactly the 16 bytes its own wave wrote; the
    // asynccnt wait is the only ordering required.

    v4f b4 = *reinterpret_cast<const v4f*>(&sbeta[tid * VEC]);
    v4f s4;
    s4.x = expf(b4.x);
    s4.y = expf(b4.y);
    s4.z = expf(b4.z);
    s4.w = expf(b4.w);

    // ---- Stream 8 (b,c) slices, reusing the scale held in registers.
    const int    jv  = blockIdx.x * TPB + tid;      // v4f index within slice
    const size_t bc0 = (size_t)blockIdx.y * SLICES; // first slice for this block

    const v4f* __restrict__ xr = reinterpret_cast<const v4f*>(x_real) + bc0 * DIMV + jv;
    const v4f* __restrict__ xi = reinterpret_cast<const v4f*>(x_imag) + bc0 * DIMV + jv;
    v4f* __restrict__ orr = reinterpret_cast<v4f*>(out) + bc0 * DIMV + jv;
    v4f* __restrict__ oim = orr + NELEMV;           // out_imag follows out_real

#pragma unroll
    for (int s = 0; s < SLICES; ++s) {
        const size_t off = (size_t)s * DIMV;
        v4f r = __builtin_nontemporal_load(&xr[off]);
        v4f m = __builtin_nontemporal_load(&xi[off]);
        r *= s4;
        m *= s4;
        __builtin_nontemporal_store(r, &orr[off]);
        __builtin_nontemporal_store(m, &oim[off]);
    }
}

extern "C" void kernel_launch(void* const* d_in, const int* in_sizes, int n_in,
                              void* d_out, int out_size, void* d_ws, size_t ws_size,
                              hipStream_t stream) {
    (void)in_sizes; (void)n_in; (void)out_size; (void)d_ws; (void)ws_size;

    const float* x_real = (const float*)d_in[0];
    const float* x_imag = (const float*)d_in[1];
    const float* betas  = (const float*)d_in[2];
    float* out = (float*)d_out;

    // x: 64 blocks cover 65536 spatial positions (256 thr * 4-wide)
    // y: 32 blocks cover 256 (b,c) slices (8 per block)
    dim3 grid(DIMV / TPB, BC / SLICES);
    diag_scale_kernel<<<grid, TPB, 0, stream>>>(x_real, x_imag, betas, out);
}